// CTCPredictionsCpu_77249281786053
// MI455X (gfx1250) — compile-verified
//
#include <hip/hip_runtime.h>
#include <hip/hip_bf16.h>
#include <stdint.h>
#include <math.h>

#define T_MAX   128
#define BATCH   64
#define NC      32
#define BW      16
#define BLANK   31          // NUM_CLASSES - 1
#define NEGV    (-1e30f)
#define NCAND   (BW + BW * NC)   // 528

// ---- CDNA5 TDM availability (device pass only) ----
#if defined(__HIP_DEVICE_COMPILE__) && defined(__gfx1250__)
# if __has_builtin(__builtin_amdgcn_tensor_load_to_lds) && __has_builtin(__builtin_amdgcn_s_wait_tensorcnt)
#  define USE_TDM 1
# endif
#endif
#ifndef USE_TDM
# define USE_TDM 0
#endif

__device__ __forceinline__ float laddexp(float a, float b) {
    // jnp.logaddexp semantics with finite -1e30 sentinels (no NaNs)
    float m = fmaxf(a, b);
    float d = fminf(a, b) - m;      // <= 0
    return m + log1pf(expf(d));
}

extern "C" __global__ __launch_bounds__(32, 1)
void ctc_beam_kernel(const float* __restrict__ data,       // [T, B, C] logits
                     const int*   __restrict__ data_len,   // [B]
                     int*         __restrict__ out)        // [B, T] int32
{
    __shared__ float logp[T_MAX][NC];          // logits -> log_softmax in place (16 KB)
    __shared__ int   pref[2][BW][T_MAX];       // double-buffered prefixes (16 KB)
    __shared__ float ext[BW][NC];
    __shared__ float cand[NCAND];
    __shared__ float lpb[BW], lpnb[BW], stayb[BW], staynb[BW], tots[BW];
    __shared__ int   lens[BW], lastc[BW], validj[BW];
    __shared__ unsigned consumed[BW];
    __shared__ int   selidx[BW];
    __shared__ int   nsrc[BW], nisx[BW], nchr[BW], nlen[BW];
    __shared__ float nlpb[BW], nlpnb[BW];

    const int lane = threadIdx.x;              // 0..31 (wave32)
    const int b    = blockIdx.x;
    const int len  = data_len[b];

    // ------------------------------------------------------------------
    // Stage 1: DMA this sample's [T=128, C=32] tile (row stride B*C) into LDS
    // via the Tensor Data Mover (single tensor_load_to_lds), else fall back.
    // ------------------------------------------------------------------
#if USE_TDM
    {
        typedef unsigned int v4u __attribute__((ext_vector_type(4)));
        typedef int          v8i __attribute__((ext_vector_type(8)));
        typedef int          v4i __attribute__((ext_vector_type(4)));

        unsigned long long ga = (unsigned long long)(uintptr_t)(data + (size_t)b * NC);
        // low 32 bits of the flat LDS pointer == byte offset within WG LDS allocation
        unsigned int lds_off = (unsigned int)(uintptr_t)(&logp[0][0]);

        // D# group 0: count=1 | lds_addr | global_addr[56:0] | type=2 (bits 127:126)
        v4u g0 = { 1u,
                   lds_off,
                   (unsigned int)(ga & 0xffffffffu),
                   (unsigned int)((ga >> 32) & 0x01ffffffu) | 0x80000000u };
        // D# group 1:
        //  d0: workgroup_mask=0, data_size=2 (4B) at bits 17:16
        //  d1: tensor_dim0 low16 at bits 63:48  -> 32<<16
        //  d2: tensor_dim1 low16 at bits 95:80  -> 128<<16
        //  d3: tile_dim0   at bits 127:112      -> 32<<16
        //  d4: tile_dim1   at bits 143:128      -> 128
        //  d5: tensor_dim0_stride low32         -> B*C = 2048 elements
        v8i g1 = { (int)0x00020000,
                   (int)(32u  << 16),
                   (int)(128u << 16),
                   (int)(32u  << 16),
                   128,
                   2048,
                   0, 0 };
        v4i g2 = { 0, 0, 0, 0 };        // 2D tensor: groups 2/3 unused
        v4i g3 = { 0, 0, 0, 0 };
        v8i g4 = { 0, 0, 0, 0, 0, 0, 0, 0 };  // extra group on 6-arg toolchain
        __builtin_amdgcn_tensor_load_to_lds(g0, g1, g2, g3, g4, 0);
        __builtin_amdgcn_s_wait_tensorcnt(0);
    }
#else
    for (int t = 0; t < T_MAX; ++t)
        logp[t][lane] = data[((size_t)t * BATCH + b) * NC + lane];
#endif
    __syncthreads();

    // ------------------------------------------------------------------
    // Stage 2: log_softmax per row, lane = class, wave-wide reductions
    // ------------------------------------------------------------------
    for (int t = 0; t < T_MAX; ++t) {
        float x = logp[t][lane];
        float m = x;
        for (int o = 16; o; o >>= 1) m = fmaxf(m, __shfl_xor(m, o, 32));
        float s = expf(x - m);
        for (int o = 16; o; o >>= 1) s += __shfl_xor(s, o, 32);
        logp[t][lane] = x - m - logf(s);
    }

    // ------------------------------------------------------------------
    // Stage 3: init beam state
    // ------------------------------------------------------------------
    if (lane < BW) {
        lens[lane] = 0;
        lpb[lane]  = (lane == 0) ? 0.0f : NEGV;
        lpnb[lane] = NEGV;
    }
    for (int i = 0; i < BW; ++i)
        for (int l = lane; l < T_MAX; l += 32)
            pref[0][i][l] = 0;
    __syncthreads();

    // ------------------------------------------------------------------
    // Stage 4: prefix beam search over valid frames
    // ------------------------------------------------------------------
    for (int t = 0; t < len; ++t) {
        const int cur = t & 1, nxt = cur ^ 1;

        // A: per-beam totals / stay candidates (lanes 0..15 = beams)
        if (lane < BW) {
            float a = lpb[lane], n = lpnb[lane];
            float to = laddexp(a, n);
            tots[lane]   = to;
            validj[lane] = (to > NEGV * 0.5f) ? 1 : 0;
            int li = lens[lane];
            int la = (li > 0) ? pref[cur][lane][li - 1] : -1;
            lastc[lane]  = la;
            stayb[lane]  = to + logp[t][BLANK];
            staynb[lane] = (li > 0) ? (n + logp[t][la]) : NEGV;
            consumed[lane] = 0u;
        }
        __syncthreads();

        // B: ext[i][c] (lane = class c)
        {
            float lp_c = logp[t][lane];
            for (int i = 0; i < BW; ++i) {
                float base = (lane == lastc[i]) ? lpb[i] : tots[i];
                ext[i][lane] = (lane == BLANK) ? NEGV : (base + lp_c);
            }
        }
        __syncthreads();

        // C: merge ext(i,c) into stay(j) when prefixes[j] == prefixes[i] ++ [c]
        if (lane < BW && validj[lane]) {
            const int j  = lane;
            const int lj = lens[j];
            float m = NEGV;
            for (int i = 0; i < BW; ++i) {
                const int li = lens[i];
                if (li + 1 != lj) continue;
                bool eq = true;
                for (int p = 0; p < li; ++p) {
                    if (pref[cur][j][p] != pref[cur][i][p]) { eq = false; break; }
                }
                if (!eq) continue;
                int c = pref[cur][j][li];
                m = laddexp(m, ext[i][c]);
                atomicOr(&consumed[i], 1u << c);
            }
            staynb[j] = laddexp(staynb[j], m);
        }
        __syncthreads();

        // D: mask consumed ext entries (lane = class)
        for (int i = 0; i < BW; ++i)
            if ((consumed[i] >> lane) & 1u) ext[i][lane] = NEGV;
        __syncthreads();

        // E: candidate score array [stay(16) | ext(16*32)]
        if (lane < BW) cand[lane] = laddexp(stayb[lane], staynb[lane]);
        for (int i = 0; i < BW; ++i) cand[BW + i * NC + lane] = ext[i][lane];
        __syncthreads();

        // F: stable top-16 (value desc, index asc) — matches jax.lax.top_k
        for (int r = 0; r < BW; ++r) {
            float bv = -3.4e38f; int bi = 0x7fffffff;
            for (int idx = lane; idx < NCAND; idx += 32) {
                float v = cand[idx];
                if (v > bv) { bv = v; bi = idx; }    // ascending scan: keeps min idx on ties
            }
            for (int o = 16; o; o >>= 1) {
                float ov = __shfl_xor(bv, o, 32);
                int   oi = __shfl_xor(bi, o, 32);
                if (ov > bv || (ov == bv && oi < bi)) { bv = ov; bi = oi; }
            }
            if (lane == 0) { selidx[r] = bi; cand[bi] = -3.4e38f; }
            __syncthreads();
        }

        // G: decode selections into new beam state (lane = new beam)
        if (lane < BW) {
            int idx = selidx[lane];
            if (idx < BW) {                          // stay
                nsrc[lane] = idx; nisx[lane] = 0; nchr[lane] = 0;
                nlen[lane] = lens[idx];
                nlpb[lane] = stayb[idx]; nlpnb[lane] = staynb[idx];
            } else {                                 // extend
                int e = idx - BW, i = e >> 5, c = e & 31;
                nsrc[lane] = i; nisx[lane] = 1; nchr[lane] = c;
                nlen[lane] = lens[i] + 1;
                nlpb[lane] = NEGV; nlpnb[lane] = ext[i][c];
            }
        }
        __syncthreads();

        // H: materialize new prefixes into the other buffer (all 32 lanes)
        for (int k = 0; k < BW; ++k) {
            const int s   = nsrc[k];
            const int bl  = lens[s];
            const int isx = nisx[k];
            const int cc  = nchr[k];
            for (int l = lane; l < T_MAX; l += 32) {
                int v = pref[cur][s][l];
                if (isx && l == bl) v = cc;
                pref[nxt][k][l] = v;
            }
        }
        __syncthreads();

        // I: commit scalar beam state
        if (lane < BW) {
            lens[lane] = nlen[lane];
            lpb[lane]  = nlpb[lane];
            lpnb[lane] = nlpnb[lane];
        }
        __syncthreads();
    }

    // ------------------------------------------------------------------
    // Stage 5: emit beam 0, padded with 0
    // ------------------------------------------------------------------
    const int fin = len & 1;
    const int L0  = lens[0];
    for (int l = lane; l < T_MAX; l += 32)
        out[b * T_MAX + l] = (l < L0) ? pref[fin][0][l] : 0;
}

extern "C" void kernel_launch(void* const* d_in, const int* in_sizes, int n_in,
                              void* d_out, int out_size, void* d_ws, size_t ws_size,
                              hipStream_t stream) {
    (void)in_sizes; (void)n_in; (void)out_size; (void)d_ws; (void)ws_size;
    const float* data = (const float*)d_in[0];
    const int*   dlen = (const int*)d_in[1];
    int*         out  = (int*)d_out;
    hipLaunchKernelGGL(ctc_beam_kernel, dim3(BATCH), dim3(32), 0, stream,
                       data, dlen, out);
}